// GCNLayer_12833362280698
// MI455X (gfx1250) — compile-verified
//
#include <hip/hip_runtime.h>

#define D 128

typedef float v2f __attribute__((ext_vector_type(2)));
typedef float v8f __attribute__((ext_vector_type(8)));

// ---------------- small utility kernels ----------------

__global__ void zero_f32(float* __restrict__ p, int n) {
    int i = blockIdx.x * blockDim.x + threadIdx.x;
    if (i < n) p[i] = 0.0f;
}

__global__ void deg_kernel(const int* __restrict__ col,
                           const float* __restrict__ w,
                           float* __restrict__ deg, int E) {
    int e = blockIdx.x * blockDim.x + threadIdx.x;
    if (e < E) unsafeAtomicAdd(&deg[col[e]], w[e]);
}

__global__ void rsqrt_kernel(const float* __restrict__ deg,
                             float* __restrict__ dis, int n) {
    int i = blockIdx.x * blockDim.x + threadIdx.x;
    if (i < n) {
        float d = deg[i];
        dis[i] = (d > 0.0f) ? __frsqrt_rn(d) : 0.0f;
    }
}

// ---------------- fused dual GEMM: hl = x@Wlin -> out, h = x@Wgcn -> ws ----------------
// Block: 512 threads = 16 waves. Each block covers 16 rows of x.
// wave 0..7  : W_lin, column tiles 0..7 (16 cols each) -> out_hl
// wave 8..15 : W_gcn, column tiles 0..7               -> h
// Uses V_WMMA_F32_16X16X4_F32, exact fp32 (matches reference numerics).

__global__ __launch_bounds__(512) void gemm_dual(const float* __restrict__ x,
                                                 const float* __restrict__ Wlin,
                                                 const float* __restrict__ Wgcn,
                                                 float* __restrict__ out_hl,
                                                 float* __restrict__ h) {
    const int row0  = blockIdx.x * 16;
    const int wave  = threadIdx.x >> 5;   // 0..15
    const int lane  = threadIdx.x & 31;
    const int which = wave >> 3;          // 0 = lin, 1 = gcn
    const int col0  = (wave & 7) * 16;

    const float* __restrict__ W   = which ? Wgcn : Wlin;
    float* __restrict__       dst = which ? h    : out_hl;

    const int m  = lane & 15;   // row within tile (A) / col within tile (B,C)
    const int hh = lane >> 4;   // lane half

    const float* __restrict__ arow = x + (size_t)(row0 + m) * D;

    v8f acc = {};
#pragma unroll
    for (int k0 = 0; k0 < D; k0 += 4) {
        // A fragment (16x4 f32): VGPR0 -> K = k0+2*hh, VGPR1 -> K = k0+2*hh+1
        v2f a;
        a.x = arow[k0 + 2 * hh + 0];
        a.y = arow[k0 + 2 * hh + 1];
        // B fragment (4x16 f32): VGPR j, lane-half hh -> row K = k0 + j + 2*hh
        v2f b;
        b.x = W[(size_t)(k0 + 2 * hh + 0) * D + col0 + m];
        b.y = W[(size_t)(k0 + 2 * hh + 1) * D + col0 + m];
        acc = __builtin_amdgcn_wmma_f32_16x16x4_f32(
            /*neg_a=*/false, a, /*neg_b=*/false, b,
            /*c_mod=*/(short)0, acc, /*reuse_a=*/false, /*reuse_b=*/false);
    }

    // C/D layout: VGPR r, lane-half hh -> row (r + 8*hh), col (col0 + m)
#pragma unroll
    for (int r = 0; r < 8; ++r) {
        dst[(size_t)(row0 + r + 8 * hh) * D + col0 + m] = acc[r];
    }
}

// ---------------- edge propagate: out[col] += h[row] * norm ----------------
// One wave (32 lanes) per edge; each lane handles a float4 chunk of the 128-dim row.

__global__ __launch_bounds__(256) void scatter_kernel(const float* __restrict__ h,
                                                      const int* __restrict__ row_idx,
                                                      const int* __restrict__ col_idx,
                                                      const float* __restrict__ w,
                                                      const float* __restrict__ dis,
                                                      float* __restrict__ out, int E) {
    const int wave = threadIdx.x >> 5;
    const int lane = threadIdx.x & 31;
    const int e    = blockIdx.x * 8 + wave;
    if (e >= E) return;

    const int r = row_idx[e];
    const int c = col_idx[e];
    const float norm = dis[r] * w[e] * dis[c];

    const float4* __restrict__ hv = (const float4*)(h + (size_t)r * D);
    float4 v = hv[lane];

    float* __restrict__ o = out + (size_t)c * D + lane * 4;
    unsafeAtomicAdd(o + 0, v.x * norm);
    unsafeAtomicAdd(o + 1, v.y * norm);
    unsafeAtomicAdd(o + 2, v.z * norm);
    unsafeAtomicAdd(o + 3, v.w * norm);
}

// ---------------- launcher ----------------

extern "C" void kernel_launch(void* const* d_in, const int* in_sizes, int n_in,
                              void* d_out, int out_size, void* d_ws, size_t ws_size,
                              hipStream_t stream) {
    const float* x    = (const float*)d_in[0];
    const int*   ei   = (const int*)d_in[1];
    const float* ew   = (const float*)d_in[2];
    const float* Wlin = (const float*)d_in[3];
    const float* Wgcn = (const float*)d_in[4];
    float*       out  = (float*)d_out;

    const int N = in_sizes[0] / D;   // 100000
    const int E = in_sizes[2];       // 1600000

    const int* row = ei;
    const int* col = ei + E;

    // workspace layout: deg[N] | dis[N] | h[N*D]
    float* deg = (float*)d_ws;
    float* dis = deg + N;
    float* h   = dis + N;   // byte offset 2*N*4 = 800000, 16B-aligned

    zero_f32<<<(N + 255) / 256, 256, 0, stream>>>(deg, N);
    deg_kernel<<<(E + 255) / 256, 256, 0, stream>>>(col, ew, deg, E);
    rsqrt_kernel<<<(N + 255) / 256, 256, 0, stream>>>(deg, dis, N);
    // N divisible by 16 (100000 = 6250*16): no row tail
    gemm_dual<<<N / 16, 512, 0, stream>>>(x, Wlin, Wgcn, out, h);
    scatter_kernel<<<(E + 7) / 8, 256, 0, stream>>>(h, row, col, ew, dis, out, E);
}